// LL_custom_70935679860990
// MI455X (gfx1250) — compile-verified
//
#include <hip/hip_runtime.h>

typedef __attribute__((ext_vector_type(2))) float v2f;
typedef __attribute__((ext_vector_type(8))) float v8f;

#define CIN    64
#define COUT   64
#define BATCH  16
#define HW     32
#define KK     576           // Cin * 3 * 3 contraction length
#define KPAD   578           // LDS row pad: breaks 64-bank conflict, keeps 8B align
#define WSTRIDE (HW * HW * KK)   // floats per output channel of weight: 589824

// ---------------------------------------------------------------------------
// Kernel 1: locally-connected layer as per-location GEMM on f32 WMMA.
// grid.x = 1024 locations (i,j); block = 128 threads = 4 waves.
// Wave w computes the 16-batch x 16-channel tile for o in [16w, 16w+16).
//   Y[16x64] = P[16x576] * W[576x64]   (K stepped by 4 per v_wmma_f32_16x16x4_f32)
// ---------------------------------------------------------------------------
__global__ __launch_bounds__(128)
void lc_gemm_kernel(const float* __restrict__ x,
                    const float* __restrict__ wgt,
                    const float* __restrict__ bias,
                    float* __restrict__ y)
{
    __shared__ float P[BATCH * KPAD];   // 36,992 B

    const int loc = blockIdx.x;          // 0..1023
    const int i   = loc >> 5;
    const int j   = loc & 31;
    const int tid = threadIdx.x;

    // Stage im2col patches: P[b][k] = x[b, c, i+p-1, j+q-1], k = c*9 + p*3 + q
    // (matches weight's (c,p,q) minor-dim flattening). x is L2-resident (4 MB).
    for (int e = tid; e < BATCH * KK; e += 128) {
        int b = e / KK;
        int k = e - b * KK;
        int c = k / 9;
        int r = k - c * 9;
        int p = r / 3;
        int q = r - p * 3;
        int h = i + p - 1;
        int w = j + q - 1;
        float v = 0.f;
        if ((unsigned)h < 32u && (unsigned)w < 32u)
            v = x[(((b * CIN) + c) * HW + h) * HW + w];
        P[b * KPAD + k] = v;
    }
    __syncthreads();

    const int wave = tid >> 5;           // wave32: 4 waves per block
    const int lane = tid & 31;
    const int n    = lane & 15;          // A row (batch m) == B col (channel n)
    const int kg   = (lane >> 4) * 2;    // K sub-offset within the 4-wide step

    // A frag source (LDS), B frag source (global weights, streamed once).
    const float* pA = &P[n * KPAD + kg];
    const float* pB = wgt + (size_t)(wave * 16 + n) * WSTRIDE
                          + (size_t)loc * KK + kg;

    v8f acc = {};
    #pragma unroll 8
    for (int k0 = 0; k0 < KK; k0 += 4) {
        v2f a = *(const v2f*)(pA + k0);                           // ds_load_b64
        v2f b = __builtin_nontemporal_load((const v2f*)(pB + k0)); // NT global_load_b64
        // 8 args: (neg_a, A, neg_b, B, c_mod, C, reuse_a, reuse_b)
        acc = __builtin_amdgcn_wmma_f32_16x16x4_f32(
            false, a, false, b, (short)0, acc, false, false);
    }

    // C/D layout: VGPR r on lane L holds Y[M = r + 8*(L>>4)][N = L&15].
    const int   o     = wave * 16 + n;
    const float bv    = bias[o * (HW * HW) + loc];
    const int   mbase = (lane >> 4) * 8;
    #pragma unroll
    for (int r = 0; r < 8; ++r) {
        int b = mbase + r;
        y[((b * COUT) + o) * (HW * HW) + loc] = acc[r] + bv;
    }
}

// ---------------------------------------------------------------------------
// Kernel 2: per-channel batch stats -> fused scale/shift.
// 64 blocks (one per channel o), 256 threads. ss[o]=gamma*invstd,
// ss[64+o]=beta-mean*scale.
// ---------------------------------------------------------------------------
__global__ __launch_bounds__(256)
void bn_stats_kernel(const float* __restrict__ y,
                     const float* __restrict__ gamma,
                     const float* __restrict__ beta,
                     float* __restrict__ ss)
{
    __shared__ float s0[256];
    __shared__ float s1[256];
    const int o = blockIdx.x;
    float sum = 0.f, sq = 0.f;
    for (int idx = threadIdx.x; idx < BATCH * HW * HW; idx += 256) {
        int b = idx >> 10;
        int t = idx & 1023;
        float v = y[b * (COUT * HW * HW) + o * (HW * HW) + t];
        sum += v;
        sq  += v * v;
    }
    s0[threadIdx.x] = sum;
    s1[threadIdx.x] = sq;
    __syncthreads();
    for (int s = 128; s > 0; s >>= 1) {
        if (threadIdx.x < s) {
            s0[threadIdx.x] += s0[threadIdx.x + s];
            s1[threadIdx.x] += s1[threadIdx.x + s];
        }
        __syncthreads();
    }
    if (threadIdx.x == 0) {
        const float inv_n = 1.0f / (float)(BATCH * HW * HW);
        float mean = s0[0] * inv_n;
        float var  = s1[0] * inv_n - mean * mean;   // biased variance
        float inv  = rsqrtf(var + 1e-5f);
        float sc   = gamma[o] * inv;
        ss[o]        = sc;
        ss[COUT + o] = beta[o] - mean * sc;
    }
}

// ---------------------------------------------------------------------------
// Kernel 3: y = relu(y*scale + shift), in place on d_out.
// ---------------------------------------------------------------------------
__global__ __launch_bounds__(256)
void bn_apply_kernel(float* __restrict__ y, const float* __restrict__ ss)
{
    int idx = blockIdx.x * 256 + threadIdx.x;
    int o = (idx >> 10) & (COUT - 1);
    float v = y[idx] * ss[o] + ss[COUT + o];
    y[idx] = v > 0.f ? v : 0.f;
}

extern "C" void kernel_launch(void* const* d_in, const int* in_sizes, int n_in,
                              void* d_out, int out_size, void* d_ws, size_t ws_size,
                              hipStream_t stream) {
    const float* x     = (const float*)d_in[0];
    const float* wgt   = (const float*)d_in[1];
    const float* bias  = (const float*)d_in[2];
    const float* gamma = (const float*)d_in[3];
    const float* beta  = (const float*)d_in[4];
    float*       out   = (float*)d_out;
    float*       ss    = (float*)d_ws;     // 128 floats

    lc_gemm_kernel<<<dim3(HW * HW), dim3(128), 0, stream>>>(x, wgt, bias, out);
    bn_stats_kernel<<<dim3(COUT), dim3(256), 0, stream>>>(out, gamma, beta, ss);
    const int total = BATCH * COUT * HW * HW;       // 1,048,576
    bn_apply_kernel<<<dim3(total / 256), dim3(256), 0, stream>>>(out, ss);
}